// SimpleAttention_50646254355060
// MI455X (gfx1250) — compile-verified
//
#include <hip/hip_runtime.h>
#include <hip/hip_bf16.h>

typedef float v2f __attribute__((ext_vector_type(2)));
typedef float v8f __attribute__((ext_vector_type(8)));

#define DIM 4096
#define N_HEADS 32
#define HEAD_DIM 128
#define MAX_SEQ 8192
#define SEQ 16
#define CACHE_POS0 (MAX_SEQ - SEQ)   // 8176: first "new" position
#define CHUNK 256                    // keys per partial-attention block
#define N_CHUNKS (MAX_SEQ / CHUNK)   // 32
#define KSEG 8                       // GEMM K-dim split factor
#define KLEN (DIM / KSEG)            // 512
#define MAT_ELEMS (SEQ * DIM)        // 65536

__device__ __forceinline__ float half16_max(float v) {
    v = fmaxf(v, __shfl_xor(v, 1, 32));
    v = fmaxf(v, __shfl_xor(v, 2, 32));
    v = fmaxf(v, __shfl_xor(v, 4, 32));
    v = fmaxf(v, __shfl_xor(v, 8, 32));
    return v;
}
__device__ __forceinline__ float half16_sum(float v) {
    v += __shfl_xor(v, 1, 32);
    v += __shfl_xor(v, 2, 32);
    v += __shfl_xor(v, 4, 32);
    v += __shfl_xor(v, 8, 32);
    return v;
}

// Partial GEMM: P[kseg] [16 x 4096] = X[16, k-seg] @ W[:, k-seg]^T
// One wave per (16x16 tile, k-segment). grid = (256, KSEG).
__global__ __launch_bounds__(32) void gemm_xwt_part_kernel(
    const float* __restrict__ X, const float* __restrict__ W, float* __restrict__ P) {
    const int lane = threadIdx.x;
    const int n0   = blockIdx.x * 16;
    const int k0   = blockIdx.y * KLEN;
    const int c16  = lane & 15;           // A: M row; B/D: N column
    const int half = lane >> 4;
    const int koff = half * 2;

    const float* xrow = X + (size_t)c16 * DIM + k0 + koff;
    const float* wrow = W + (size_t)(n0 + c16) * DIM + k0 + koff;

    v8f c0 = {}, c1 = {}, c2 = {}, c3 = {};
    for (int k = 0; k < KLEN; k += 64) {
        __builtin_prefetch(wrow + k + 128, 0, 1);   // global_prefetch_b8, ~512B ahead
#pragma unroll
        for (int kk = 0; kk < 64; kk += 16) {
            const int kb = k + kk;
            v2f a0 = *(const v2f*)(xrow + kb);
            v2f b0 = *(const v2f*)(wrow + kb);
            c0 = __builtin_amdgcn_wmma_f32_16x16x4_f32(false, a0, false, b0, (short)0, c0, false, false);
            v2f a1 = *(const v2f*)(xrow + kb + 4);
            v2f b1 = *(const v2f*)(wrow + kb + 4);
            c1 = __builtin_amdgcn_wmma_f32_16x16x4_f32(false, a1, false, b1, (short)0, c1, false, false);
            v2f a2 = *(const v2f*)(xrow + kb + 8);
            v2f b2 = *(const v2f*)(wrow + kb + 8);
            c2 = __builtin_amdgcn_wmma_f32_16x16x4_f32(false, a2, false, b2, (short)0, c2, false, false);
            v2f a3 = *(const v2f*)(xrow + kb + 12);
            v2f b3 = *(const v2f*)(wrow + kb + 12);
            c3 = __builtin_amdgcn_wmma_f32_16x16x4_f32(false, a3, false, b3, (short)0, c3, false, false);
        }
    }
    float* out = P + (size_t)blockIdx.y * MAT_ELEMS;
#pragma unroll
    for (int i = 0; i < 8; ++i) {
        const int row = i + half * 8;
        out[(size_t)row * DIM + n0 + c16] = (c0[i] + c1[i]) + (c2[i] + c3[i]);
    }
}

// Deterministic K-segment reduction: Y[i] = sum_s P[s][i], i in [0, 65536).
__global__ __launch_bounds__(256) void ksum_kernel(
    const float* __restrict__ P, float* __restrict__ Y) {
    const int i = blockIdx.x * 256 + threadIdx.x;
    float s = 0.0f;
#pragma unroll
    for (int c = 0; c < KSEG; ++c) s += P[(size_t)c * MAT_ELEMS + i];
    Y[i] = s;
}

// Flash-attention partial over one (head, key-chunk). One wave per block.
// gridDim.x = N_HEADS * N_CHUNKS = 1024.
__global__ __launch_bounds__(32) void attn_partial_kernel(
    const float* __restrict__ q_ws,   // [16, 4096] : Q, head h at col h*128
    const float* __restrict__ k_new,  // [16, 4096] : new K rows (positions 8176..8191)
    const float* __restrict__ v_new,  // [16, 4096] : new V rows
    const float* __restrict__ k_cache,// [32, 8192, 128]
    const float* __restrict__ v_cache,// [32, 8192, 128]
    float* __restrict__ part_o,       // [1024][16][128]
    float* __restrict__ part_m,       // [1024][16]
    float* __restrict__ part_l) {     // [1024][16]
    const int blk   = blockIdx.x;
    const int h     = blk / N_CHUNKS;
    const int chnk  = blk % N_CHUNKS;
    const int base0 = chnk * CHUNK;
    const int lane  = threadIdx.x;
    const int c16   = lane & 15;
    const int half  = lane >> 4;
    const int koff  = half * 2;

    __shared__ float p_lds[16 * 17];

    // Preload Q fragments for the full K=128 reduction (32 steps of K=4).
    v2f qf[32];
    const float* qrow = q_ws + (size_t)c16 * DIM + h * HEAD_DIM + koff;
#pragma unroll
    for (int i = 0; i < 32; ++i) qf[i] = *(const v2f*)(qrow + 4 * i);

    float mstat[8], lstat[8];
    v8f acc[8];
#pragma unroll
    for (int i = 0; i < 8; ++i) { mstat[i] = -__builtin_inff(); lstat[i] = 0.0f; acc[i] = {}; }

    const float scale = 0.088388347648318447f;  // 1/sqrt(128)
    const size_t hoff = (size_t)h * MAX_SEQ * HEAD_DIM;

    for (int t = 0; t < CHUNK / 16; ++t) {
        const int jbase = base0 + t * 16;

        // ---- S tile = Q @ K^T for 16 keys; lane supplies key row jbase+c16 ----
        const int jk = jbase + c16;
        const float* krow = (jk < CACHE_POS0)
            ? (k_cache + hoff + (size_t)jk * HEAD_DIM + koff)
            : (k_new + (size_t)(jk - CACHE_POS0) * DIM + h * HEAD_DIM + koff);
        __builtin_prefetch(krow + 16 * HEAD_DIM, 0, 1);   // next tile's key row
        v8f s0 = {}, s1 = {}, s2 = {}, s3 = {};
#pragma unroll
        for (int i = 0; i < 32; i += 4) {
            v2f b0 = *(const v2f*)(krow + 4 * i);
            s0 = __builtin_amdgcn_wmma_f32_16x16x4_f32(false, qf[i],     false, b0, (short)0, s0, false, false);
            v2f b1 = *(const v2f*)(krow + 4 * i + 4);
            s1 = __builtin_amdgcn_wmma_f32_16x16x4_f32(false, qf[i + 1], false, b1, (short)0, s1, false, false);
            v2f b2 = *(const v2f*)(krow + 4 * i + 8);
            s2 = __builtin_amdgcn_wmma_f32_16x16x4_f32(false, qf[i + 2], false, b2, (short)0, s2, false, false);
            v2f b3 = *(const v2f*)(krow + 4 * i + 12);
            s3 = __builtin_amdgcn_wmma_f32_16x16x4_f32(false, qf[i + 3], false, b3, (short)0, s3, false, false);
        }

        // ---- scale, causal mask, online softmax (row stats per VGPR slot) ----
#pragma unroll
        for (int i = 0; i < 8; ++i) {
            const int row = i + half * 8;            // query index
            float sv = ((s0[i] + s1[i]) + (s2[i] + s3[i])) * scale;
            if (jbase + c16 > CACHE_POS0 + row) sv = -1e30f;  // causal mask on new keys
            const float tmax = half16_max(sv);
            const float mnew = fmaxf(mstat[i], tmax);
            const float alpha = __expf(mstat[i] - mnew);
            const float p = __expf(sv - mnew);
            const float psum = half16_sum(p);
            lstat[i] = lstat[i] * alpha + psum;
            mstat[i] = mnew;
#pragma unroll
            for (int nt = 0; nt < 8; ++nt) acc[nt][i] *= alpha;
            p_lds[row * 17 + c16] = p;               // D-layout -> LDS (row=query, col=key)
        }
        __syncthreads();

        // ---- reload P in A-matrix layout (M=query, K=key) ----
        v2f pf[4];
#pragma unroll
        for (int kk = 0; kk < 4; ++kk) {
            pf[kk].x = p_lds[c16 * 17 + 4 * kk + koff];
            pf[kk].y = p_lds[c16 * 17 + 4 * kk + koff + 1];
        }
        __syncthreads();

        // ---- acc += P @ V ; B rows are V key-rows, columns are head dims ----
#pragma unroll
        for (int kk = 0; kk < 4; ++kk) {
            const int r0 = jbase + 4 * kk + koff;
            const int r1 = r0 + 1;
            const float* vr0 = (r0 < CACHE_POS0)
                ? (v_cache + hoff + (size_t)r0 * HEAD_DIM)
                : (v_new + (size_t)(r0 - CACHE_POS0) * DIM + h * HEAD_DIM);
            const float* vr1 = (r1 < CACHE_POS0)
                ? (v_cache + hoff + (size_t)r1 * HEAD_DIM)
                : (v_new + (size_t)(r1 - CACHE_POS0) * DIM + h * HEAD_DIM);
#pragma unroll
            for (int nt = 0; nt < 8; ++nt) {
                v2f b;
                b.x = vr0[nt * 16 + c16];
                b.y = vr1[nt * 16 + c16];
                acc[nt] = __builtin_amdgcn_wmma_f32_16x16x4_f32(false, pf[kk], false, b, (short)0, acc[nt], false, false);
            }
        }
    }

    // ---- write partials ----
    float* po = part_o + (size_t)blk * (16 * HEAD_DIM);
#pragma unroll
    for (int i = 0; i < 8; ++i) {
        const int row = i + half * 8;
#pragma unroll
        for (int nt = 0; nt < 8; ++nt)
            po[row * HEAD_DIM + nt * 16 + c16] = acc[nt][i];
    }
    if (c16 == 0) {
#pragma unroll
        for (int i = 0; i < 8; ++i) {
            const int row = i + half * 8;
            part_m[blk * 16 + row] = mstat[i];
            part_l[blk * 16 + row] = lstat[i];
        }
    }
}

// Merge the N_CHUNKS partials of one head. gridDim.x = 32 heads, block = 128 (dim).
__global__ __launch_bounds__(128) void attn_reduce_kernel(
    const float* __restrict__ part_o, const float* __restrict__ part_m,
    const float* __restrict__ part_l, float* __restrict__ attn_out) {
    const int h = blockIdx.x;
    const int d = threadIdx.x;
    for (int row = 0; row < SEQ; ++row) {
        float M = -__builtin_inff();
        for (int c = 0; c < N_CHUNKS; ++c)
            M = fmaxf(M, part_m[(h * N_CHUNKS + c) * 16 + row]);
        float L = 0.0f, o = 0.0f;
        for (int c = 0; c < N_CHUNKS; ++c) {
            const int b = h * N_CHUNKS + c;
            const float w = __expf(part_m[b * 16 + row] - M);
            L += part_l[b * 16 + row] * w;
            o += part_o[(size_t)b * (16 * HEAD_DIM) + row * HEAD_DIM + d] * w;
        }
        attn_out[(size_t)row * DIM + h * HEAD_DIM + d] = o / L;
    }
}

extern "C" void kernel_launch(void* const* d_in, const int* in_sizes, int n_in,
                              void* d_out, int out_size, void* d_ws, size_t ws_size,
                              hipStream_t stream) {
    const float* x       = (const float*)d_in[0];
    const float* wq      = (const float*)d_in[1];
    const float* wk      = (const float*)d_in[2];
    const float* wv      = (const float*)d_in[3];
    const float* wo      = (const float*)d_in[4];
    const float* k_cache = (const float*)d_in[5];
    const float* v_cache = (const float*)d_in[6];
    // d_in[7] = input_pos (static: positions 8176..8191 by construction)

    float* ws      = (float*)d_ws;
    float* q_ws    = ws;                                   // 65536
    float* k_ws    = q_ws + MAT_ELEMS;                     // 65536
    float* v_ws    = k_ws + MAT_ELEMS;                     // 65536
    float* attn_ws = v_ws + MAT_ELEMS;                     // 65536
    float* part_o  = attn_ws + MAT_ELEMS;                  // 1024*2048 (8 MB)
    float* part_m  = part_o + (size_t)N_HEADS * N_CHUNKS * 16 * HEAD_DIM;
    float* part_l  = part_m + N_HEADS * N_CHUNKS * 16;
    float* q_part  = part_l + N_HEADS * N_CHUNKS * 16;     // KSEG*65536 each
    float* k_part  = q_part + (size_t)KSEG * MAT_ELEMS;
    float* v_part  = k_part + (size_t)KSEG * MAT_ELEMS;
    float* o_part  = q_part;                               // reuse (QKV parts dead by then)

    const dim3 gpart(DIM / 16, KSEG);   // 256 x 8 = 2048 waves per matrix
    const dim3 gsum(MAT_ELEMS / 256);   // 256 blocks

    gemm_xwt_part_kernel<<<gpart, 32, 0, stream>>>(x, wq, q_part);
    gemm_xwt_part_kernel<<<gpart, 32, 0, stream>>>(x, wk, k_part);
    gemm_xwt_part_kernel<<<gpart, 32, 0, stream>>>(x, wv, v_part);
    ksum_kernel<<<gsum, 256, 0, stream>>>(q_part, q_ws);
    ksum_kernel<<<gsum, 256, 0, stream>>>(k_part, k_ws);
    ksum_kernel<<<gsum, 256, 0, stream>>>(v_part, v_ws);

    attn_partial_kernel<<<N_HEADS * N_CHUNKS, 32, 0, stream>>>(
        q_ws, k_ws, v_ws, k_cache, v_cache, part_o, part_m, part_l);

    attn_reduce_kernel<<<N_HEADS, 128, 0, stream>>>(part_o, part_m, part_l, attn_ws);

    gemm_xwt_part_kernel<<<gpart, 32, 0, stream>>>(attn_ws, wo, o_part);
    ksum_kernel<<<gsum, 256, 0, stream>>>(o_part, (float*)d_out);
}